// NGCF_38371237823058
// MI455X (gfx1250) — compile-verified
//
#include <hip/hip_runtime.h>

#define N_USERS 40000
#define N_ITEMS 60000
#define N_NODES (N_USERS + N_ITEMS)
#define EMBED_DIM 64
#define NNZ 1200000
#define BATCH 1024
#define NEG_SLOPE 0.2f
#define L2EPS 1e-12f
#define EALL_STRIDE 256   // 4 layers * 64 columns

typedef __attribute__((ext_vector_type(2))) float v2f;
typedef __attribute__((ext_vector_type(8))) float v8f;

__device__ __forceinline__ v8f wmma_f32_16x16x4(v2f a, v2f b, v8f c) {
  // 8 args: (neg_a, A, neg_b, B, c_mod, C, reuse_a, reuse_b)
  return __builtin_amdgcn_wmma_f32_16x16x4_f32(false, a, false, b, (short)0, c,
                                               false, false);
}

// ---------------------------------------------------------------------------
// E_all[:, 0:64] = concat(user_embed, item_embed)
// ---------------------------------------------------------------------------
__global__ void k_init(const float* __restrict__ ue, const float* __restrict__ ie,
                       float* __restrict__ Eall) {
  long long i = (long long)blockIdx.x * blockDim.x + threadIdx.x;
  if (i >= (long long)N_NODES * EMBED_DIM) return;
  int n = (int)(i >> 6);
  int d = (int)(i & 63);
  float v = (n < N_USERS) ? ue[(size_t)n * EMBED_DIM + d]
                          : ie[(size_t)(n - N_USERS) * EMBED_DIM + d];
  Eall[(size_t)n * EALL_STRIDE + d] = v;
}

__global__ void k_zero(float* __restrict__ p, long long n) {
  long long i = (long long)blockIdx.x * blockDim.x + threadIdx.x;
  if (i < n) p[i] = 0.0f;
}

// ---------------------------------------------------------------------------
// SpMM scatter: E_gc[rows[e]] += vals[e] * E_all[cols[e], colOff:colOff+64]
// One lane handles 2 dims of one edge (32 lanes = 64 dims).
// ---------------------------------------------------------------------------
__global__ void k_spmm(const float* __restrict__ vals, const int* __restrict__ rows,
                       const int* __restrict__ cols, const float* __restrict__ Eall,
                       int colOff, float* __restrict__ Egc) {
  long long t = (long long)blockIdx.x * blockDim.x + threadIdx.x;
  long long e = t >> 5;
  if (e >= (long long)NNZ) return;
  int d = ((int)t & 31) * 2;
  float v = vals[e];
  int r = rows[e];
  int c = cols[e];
  float2 s = *(const float2*)(Eall + (size_t)c * EALL_STRIDE + colOff + d);
  float* dst = Egc + (size_t)r * EMBED_DIM + d;
  atomicAdd(dst, v * s.x);
  atomicAdd(dst + 1, v * s.y);
}

// ---------------------------------------------------------------------------
// Layer transform: for 16-row tile:
//   D = (E+Egc)@W1 + (E*Egc)@W2 ; leaky_relu ; L2-normalize rows
// One wave per tile, fp32 WMMA 16x16x4, K=64 -> 16 steps, 4 N-tiles.
// ---------------------------------------------------------------------------
__global__ void __launch_bounds__(128) k_transform(
    float* __restrict__ Eall, const float* __restrict__ Egc,
    const float* __restrict__ W1, const float* __restrict__ W2,
    int colIn, int colOut) {
  int wave = threadIdx.x >> 5;
  int lane = threadIdx.x & 31;
  int tile = blockIdx.x * 4 + wave;
  if (tile >= N_NODES / 16) return;          // wave-uniform

  int row0 = tile * 16;
  int m    = lane & 15;                      // row within tile (A), col within tile (B)
  int hi   = lane >> 4;                      // half-wave select
  int koff = hi * 2;                         // K offset within group of 4

  const float* eRow = Eall + (size_t)(row0 + m) * EALL_STRIDE + colIn;
  const float* gRow = Egc + (size_t)(row0 + m) * EMBED_DIM;

  // Prebuild A fragments for all 16 K-steps: a1 = E+Egc, a2 = E*Egc
  v2f a1[16], a2[16];
#pragma unroll
  for (int k = 0; k < 16; ++k) {
    float2 e = *(const float2*)(eRow + 4 * k + koff);
    float2 g = *(const float2*)(gRow + 4 * k + koff);
    v2f t1; t1.x = e.x + g.x; t1.y = e.y + g.y;
    v2f t2; t2.x = e.x * g.x; t2.y = e.y * g.y;
    a1[k] = t1;
    a2[k] = t2;
  }

  v8f acc[4];
#pragma unroll
  for (int nt = 0; nt < 4; ++nt) {
    v8f c = {0.f, 0.f, 0.f, 0.f, 0.f, 0.f, 0.f, 0.f};
    int col = nt * 16 + m;
#pragma unroll
    for (int k = 0; k < 16; ++k) {
      int kr = 4 * k + koff;
      v2f b1; b1.x = W1[(size_t)kr * 64 + col]; b1.y = W1[(size_t)(kr + 1) * 64 + col];
      v2f b2; b2.x = W2[(size_t)kr * 64 + col]; b2.y = W2[(size_t)(kr + 1) * 64 + col];
      c = wmma_f32_16x16x4(a1[k], b1, c);
      c = wmma_f32_16x16x4(a2[k], b2, c);
    }
    acc[nt] = c;
  }

  // Leaky ReLU
#pragma unroll
  for (int nt = 0; nt < 4; ++nt) {
#pragma unroll
    for (int j = 0; j < 8; ++j) {
      float v = acc[nt][j];
      acc[nt][j] = (v >= 0.f) ? v : NEG_SLOPE * v;
    }
  }

  // Row L2 normalize: row (j + hi*8) lives entirely in this 16-lane half for VGPR j
#pragma unroll
  for (int j = 0; j < 8; ++j) {
    float s = 0.f;
#pragma unroll
    for (int nt = 0; nt < 4; ++nt) s += acc[nt][j] * acc[nt][j];
    s += __shfl_xor(s, 1, 32);
    s += __shfl_xor(s, 2, 32);
    s += __shfl_xor(s, 4, 32);
    s += __shfl_xor(s, 8, 32);
    float inv = 1.0f / fmaxf(sqrtf(s), L2EPS);
#pragma unroll
    for (int nt = 0; nt < 4; ++nt) acc[nt][j] *= inv;
  }

  // Store to E_all[:, colOut:colOut+64]
  float* outBase = Eall + colOut;
#pragma unroll
  for (int nt = 0; nt < 4; ++nt) {
#pragma unroll
    for (int j = 0; j < 8; ++j) {
      int r = row0 + j + hi * 8;
      outBase[(size_t)r * EALL_STRIDE + nt * 16 + m] = acc[nt][j];
    }
  }
}

// ---------------------------------------------------------------------------
// Score: out[B,B] = E_all[user_ids] @ E_all[N_USERS + item_ids]^T, K = 256
// One wave per 16x16 output tile; 64x64 tiles; 64 fp32 WMMA K-steps.
// ---------------------------------------------------------------------------
__global__ void __launch_bounds__(256) k_score(const float* __restrict__ Eall,
                                               const int* __restrict__ uids,
                                               const int* __restrict__ iids,
                                               float* __restrict__ out) {
  int wave = threadIdx.x >> 5;
  int lane = threadIdx.x & 31;
  int tile = blockIdx.x * 8 + wave;          // 512 blocks * 8 waves = 4096 tiles
  int mt = tile >> 6;
  int nt = tile & 63;
  int m    = lane & 15;
  int hi   = lane >> 4;
  int koff = hi * 2;

  const float* uRow = Eall + (size_t)uids[mt * 16 + m] * EALL_STRIDE;
  const float* iRow = Eall + ((size_t)iids[nt * 16 + m] + N_USERS) * EALL_STRIDE;

  v8f c = {0.f, 0.f, 0.f, 0.f, 0.f, 0.f, 0.f, 0.f};
#pragma unroll
  for (int k = 0; k < 64; ++k) {
    float2 af = *(const float2*)(uRow + 4 * k + koff);
    float2 bf = *(const float2*)(iRow + 4 * k + koff);
    v2f a; a.x = af.x; a.y = af.y;
    v2f b; b.x = bf.x; b.y = bf.y;
    c = wmma_f32_16x16x4(a, b, c);
  }

#pragma unroll
  for (int j = 0; j < 8; ++j) {
    int row = mt * 16 + j + hi * 8;
    out[(size_t)row * BATCH + nt * 16 + m] = c[j];
  }
}

// ---------------------------------------------------------------------------
extern "C" void kernel_launch(void* const* d_in, const int* in_sizes, int n_in,
                              void* d_out, int out_size, void* d_ws, size_t ws_size,
                              hipStream_t stream) {
  const float* ue = (const float*)d_in[0];
  const float* ie = (const float*)d_in[1];
  const float* W1s[3] = {(const float*)d_in[2], (const float*)d_in[3], (const float*)d_in[4]};
  const float* W2s[3] = {(const float*)d_in[5], (const float*)d_in[6], (const float*)d_in[7]};
  const float* vals = (const float*)d_in[8];
  const int* rows = (const int*)d_in[9];
  const int* cols = (const int*)d_in[10];
  const int* uids = (const int*)d_in[11];
  const int* iids = (const int*)d_in[12];
  float* out = (float*)d_out;

  // Workspace layout: E_all [N_NODES x 256] fp32, then E_gc [N_NODES x 64] fp32
  float* Eall = (float*)d_ws;
  float* Egc  = (float*)((char*)d_ws +
                         (size_t)N_NODES * EALL_STRIDE * sizeof(float));

  dim3 b256(256);

  {
    long long n = (long long)N_NODES * EMBED_DIM;
    k_init<<<(unsigned)((n + 255) / 256), b256, 0, stream>>>(ue, ie, Eall);
  }

  for (int l = 0; l < 3; ++l) {
    long long ng = (long long)N_NODES * EMBED_DIM;
    k_zero<<<(unsigned)((ng + 255) / 256), b256, 0, stream>>>(Egc, ng);

    long long nth = (long long)NNZ * 32;
    k_spmm<<<(unsigned)((nth + 255) / 256), b256, 0, stream>>>(
        vals, rows, cols, Eall, l * 64, Egc);

    int tiles = N_NODES / 16;  // 6250
    k_transform<<<(tiles + 3) / 4, dim3(128), 0, stream>>>(
        Eall, Egc, W1s[l], W2s[l], l * 64, (l + 1) * 64);
  }

  k_score<<<512, b256, 0, stream>>>(Eall, uids, iids, out);
}